// WordSelectionHead_69836168233628
// MI455X (gfx1250) — compile-verified
//
#include <hip/hip_runtime.h>
#include <math.h>

// Problem constants (match reference)
static constexpr int Bb = 4;
static constexpr int Nn = 192;
static constexpr int Hh = 768;
static constexpr int Ll = 9;
static constexpr int ROWS = Bb * Nn;   // 768 == Hh (coincidence, kept separate)

typedef __attribute__((ext_vector_type(2))) float v2f;
typedef __attribute__((ext_vector_type(8))) float v8f;
typedef __attribute__((ext_vector_type(4))) unsigned int v4u;
typedef __attribute__((ext_vector_type(4))) int v4i;
typedef __attribute__((ext_vector_type(8))) int v8i;

// ---- CDNA5 data-movement feature detection (compile-safe cascade) ----------
#if __has_builtin(__builtin_amdgcn_tensor_load_to_lds) && __has_builtin(__builtin_amdgcn_s_wait_tensorcnt)
#define STAGE_TDM 1
#elif __has_builtin(__builtin_amdgcn_global_load_async_to_lds_b128)
#define STAGE_ASYNC 1
#endif

typedef __attribute__((address_space(3))) void lds_void_t;
typedef __attribute__((address_space(1))) const void gbl_cvoid_t;

__device__ __forceinline__ unsigned lds_addr_of(void* p) {
    // AS3 pointers are 32-bit LDS byte offsets
    return (unsigned)(size_t)(lds_void_t*)p;
}

__device__ __forceinline__ float gelu_exact(float x) {
    // torch nn.GELU(approximate='none'): 0.5*x*(1+erf(x/sqrt(2)))
    return 0.5f * x * (1.0f + erff(x * 0.70710678118654752440f));
}

// ---------------------------------------------------------------------------
// Kernel 1: HS = pooled @ Ws^T + bs ; HT = pooled @ Wt^T + bt
// One wave computes one 16x16 fp32 tile via V_WMMA_F32_16X16X4_F32.
// A 16x4 f32 layout: lanes 0-15 -> M=lane, K={0,1}; lanes 16-31 -> M=lane-16, K={2,3}
// B 4x16 f32 layout mirrors A striping: per lane a float2 of 2 consecutive K values.
// ---------------------------------------------------------------------------
__global__ void __launch_bounds__(32)
proj_gemm_kernel(const float* __restrict__ pooled,
                 const float* __restrict__ Ws, const float* __restrict__ bs,
                 const float* __restrict__ Wt, const float* __restrict__ bt,
                 float* __restrict__ HS, float* __restrict__ HT)
{
    const int rowBase = blockIdx.x * 16;     // row tile of pooled (B*N rows)
    const int colBase = blockIdx.y * 16;     // output-feature tile
    const int which   = blockIdx.z;          // 0 = source, 1 = target

    const float* __restrict__ W  = which ? Wt : Ws;
    const float* __restrict__ bv = which ? bt : bs;
    float* __restrict__ Out      = which ? HT : HS;

    const int lane = threadIdx.x & 31;
    const int m    = lane & 15;
    const int hi   = lane >> 4;              // 0: K-pair {0,1}, 1: K-pair {2,3}
    const int off  = hi << 1;

    const float* aRow = pooled + (size_t)(rowBase + m) * Hh + off;   // A[M=m, K=k0+off..]
    const float* bRow = W      + (size_t)(colBase + m) * Hh + off;   // B[K, N=m] = W[col, h]

    v8f c = {};
    for (int k0 = 0; k0 < Hh; k0 += 4) {
        v2f a = *(const v2f*)(aRow + k0);
        v2f b = *(const v2f*)(bRow + k0);
        c = __builtin_amdgcn_wmma_f32_16x16x4_f32(
                /*neg_a=*/false, a, /*neg_b=*/false, b,
                /*c_mod=*/(short)0, c, /*reuse_a=*/false, /*reuse_b=*/false);
    }

    // D layout: VGPR j -> M = hi*8 + j, N = m
    const float bias = bv[colBase + m];
    float* ocol = Out + (size_t)colBase + m;
    #pragma unroll
    for (int j = 0; j < 8; ++j) {
        ocol[(size_t)(rowBase + hi * 8 + j) * Hh] = c[j] + bias;
    }
}

// ---------------------------------------------------------------------------
// Kernel 2: fused  out[b,s,t,l] = sum_h gelu(HS[b,s,h] + HT[b,t,h]) * Wo[l,h]
// Block = 512 threads (16 waves), 16(s) x 16(t) tile for one batch b.
// HT tile (16 rows x 768 fp32 = 48KB) staged into LDS by the Tensor Data Mover
// (one tensor_load_to_lds per block, TENSORcnt-tracked), reused by all 16 waves.
// Never materializes the 453 MB hidden tensor.
// ---------------------------------------------------------------------------
__global__ void __launch_bounds__(512)
pair_gelu_proj_kernel(const float* __restrict__ HS, const float* __restrict__ HT,
                      const float* __restrict__ Wo, float* __restrict__ out)
{
    __shared__ float s_ht[16 * Hh];          // 49,152 bytes

    const int b     = blockIdx.z;
    const int sBase = blockIdx.y * 16;
    const int tBase = blockIdx.x * 16;

    const float* src = HT + (size_t)(b * Nn + tBase) * Hh;   // 16 contiguous rows

#if defined(STAGE_TDM)
    // ---- Tensor Data Mover: one DMA for the whole 48KB tile (wave 0 only) ----
    if (threadIdx.x < 32) {
        const unsigned long long ga = (unsigned long long)(const void*)src;
        const unsigned lds = lds_addr_of((void*)s_ht);

        // D# group 0 (ISA 8.3): count=1 | lds_addr | global_addr[56:0] | type=2
        v4u g0 = { 1u,
                   lds,
                   (unsigned)(ga & 0xFFFFFFFFu),
                   (unsigned)((ga >> 32) & 0x01FFFFFFu) | (2u << 30) };

        // D# group 1 (ISA 8.4): 2-D tile, data_size=4B,
        // tensor_dim0 = 768 (row length), tensor_dim1 = 16 (rows),
        // tile_dim0 = 768, tile_dim1 = 16, tensor_dim0_stride = 768.
        v8i g1 = { (int)(2u << 16),              // wg_mask=0 | data_size=2 (4B)
                   (int)(((unsigned)Hh) << 16),  // tensor_dim0[15:0] @ bits 63:48
                   (int)(16u << 16),             // tensor_dim0 hi=0 | tensor_dim1=16
                   (int)(((unsigned)Hh) << 16),  // tensor_dim1 hi=0 | tile_dim0=768
                   16,                           // tile_dim1=16 | tile_dim2=0
                   Hh,                           // tensor_dim0_stride[31:0]
                   0, 0 };

        v4i g2 = { 0, 0, 0, 0 };                 // 2-D: groups 2/3 unused
        v4i g3 = { 0, 0, 0, 0 };

#if __clang_major__ >= 23
        v8i gx = { 0, 0, 0, 0, 0, 0, 0, 0 };
        __builtin_amdgcn_tensor_load_to_lds(g0, g1, g2, g3, gx, 0);
#else
        __builtin_amdgcn_tensor_load_to_lds(g0, g1, g2, g3, 0);
#endif
        __builtin_amdgcn_s_wait_tensorcnt(0);
    }
    __syncthreads();
#elif defined(STAGE_ASYNC)
    // ---- Async global->LDS copies through ASYNCcnt ----
    {
        const float4* s4 = (const float4*)src;
        float4* d4 = (float4*)s_ht;
        for (int i = threadIdx.x; i < (16 * Hh) / 4; i += 512) {
            __builtin_amdgcn_global_load_async_to_lds_b128(
                (gbl_cvoid_t*)(s4 + i), (lds_void_t*)(d4 + i), 0, 0);
        }
#if __has_builtin(__builtin_amdgcn_s_wait_asynccnt)
        __builtin_amdgcn_s_wait_asynccnt(0);
#else
        asm volatile("s_wait_asynccnt 0" ::: "memory");
#endif
    }
    __syncthreads();
#else
    // ---- Plain staged copy fallback ----
    {
        const float4* s4 = (const float4*)src;
        float4* d4 = (float4*)s_ht;
        for (int i = threadIdx.x; i < (16 * Hh) / 4; i += 512)
            d4[i] = s4[i];
    }
    __syncthreads();
#endif

    const int wave = threadIdx.x >> 5;       // 0..15 -> s row within tile
    const int lane = threadIdx.x & 31;
    const int m    = lane & 15;              // t_local (A) / l (B)
    const int hi   = lane >> 4;
    const int off  = hi << 1;

    const float* hsRow = HS   + (size_t)(b * Nn + sBase + wave) * Hh + off;
    const float* htRow = s_ht + (size_t)m * Hh + off;                 // LDS
    const float* woRow = Wo   + (size_t)((m < Ll) ? m : 0) * Hh + off;
    const float  wsc   = (m < Ll) ? 1.0f : 0.0f;   // zero B rows l>=9, EXEC stays full

    __builtin_prefetch(hsRow, 0, 3);         // global_prefetch_b8
    __builtin_prefetch(woRow, 0, 3);

    v8f c = {};
    for (int k0 = 0; k0 < Hh; k0 += 4) {
        v2f hsv = *(const v2f*)(hsRow + k0);
        v2f htv = *(const v2f*)(htRow + k0);   // ds_load_b64
        v2f wov = *(const v2f*)(woRow + k0);

        v2f a, bm;
        a.x  = gelu_exact(hsv.x + htv.x);
        a.y  = gelu_exact(hsv.y + htv.y);
        bm.x = wov.x * wsc;
        bm.y = wov.y * wsc;

        c = __builtin_amdgcn_wmma_f32_16x16x4_f32(
                false, a, false, bm, (short)0, c, false, false);
    }

    // D layout: VGPR j -> M (t_local) = hi*8 + j, N (l) = m
    if (m < Ll) {
        float* obase = out + (((size_t)(b * Nn + sBase + wave) * Nn) + tBase) * Ll + m;
        #pragma unroll
        for (int j = 0; j < 8; ++j) {
            obase[(size_t)(hi * 8 + j) * Ll] = c[j];
        }
    }
}

// ---------------------------------------------------------------------------
extern "C" void kernel_launch(void* const* d_in, const int* in_sizes, int n_in,
                              void* d_out, int out_size, void* d_ws, size_t ws_size,
                              hipStream_t stream) {
    const float* pooled = (const float*)d_in[0];
    const float* Ws     = (const float*)d_in[1];
    const float* bs     = (const float*)d_in[2];
    const float* Wt     = (const float*)d_in[3];
    const float* bt     = (const float*)d_in[4];
    const float* Wo     = (const float*)d_in[5];

    float* HS  = (float*)d_ws;                       // [768 x 768]
    float* HT  = HS + (size_t)ROWS * Hh;             // [768 x 768]
    float* out = (float*)d_out;                      // [4,192,192,9]

    dim3 g1(ROWS / 16, Hh / 16, 2);                  // 48 x 48 x 2 tiles
    proj_gemm_kernel<<<g1, 32, 0, stream>>>(pooled, Ws, bs, Wt, bt, HS, HT);

    dim3 g2(Nn / 16, Nn / 16, Bb);                   // 12 x 12 x 4 blocks
    pair_gelu_proj_kernel<<<g2, 512, 0, stream>>>(HS, HT, Wo, out);
}